// TweetRep_1537598292345
// MI455X (gfx1250) — compile-verified
//
#include <hip/hip_runtime.h>
#include <hip/hip_bf16.h>

// One wave (32 lanes) per "tweet" (L=30 tokens, E=64 dims).
// Block = 256 threads = 8 waves = 8 tweets; H*W = 1024 divisible by 8 so all
// 8 tweets of a block share the same (b,t) output plane -> coalesced output
// stores via an LDS transpose.
//
// Scores b = Wg @ V with V = sum_l Wg[l]  ==  rowsum(Wg @ Wg^T): computed as a
// 32x32 Gram matrix over K=64 with V_WMMA_F32_16X16X4_F32 (full f32 precision,
// 64 WMMAs per wave). Padding rows 30,31 are NOT zeroed: junk only reaches
// G[:,30:32] (masked in the row-sum via a free fma) and G[30:32,:] (never read).

typedef float v2f __attribute__((ext_vector_type(2)));
typedef float v4f __attribute__((ext_vector_type(4)));
typedef float v8f __attribute__((ext_vector_type(8)));

#define NWAVES 8
#define LTOK   30
#define EDIM   64
#define ROWSTR 66   // padded row stride (floats) to dodge LDS bank conflicts

__global__ __launch_bounds__(256) void tweetrep_kernel(
    const int*   __restrict__ x,     // (49152, 30) token indices
    const float* __restrict__ emb,   // (50000, 64) f32
    float*       __restrict__ out)   // (B, T*64, 32, 32) f32
{
    __shared__ float wg[NWAVES][LTOK][ROWSTR];  // 63360 B
    __shared__ int   sIdx[NWAVES][32];          //  1024 B
    __shared__ float bArr[NWAVES][32];          //  1024 B

    const int tid   = threadIdx.x;
    const int wave  = tid >> 5;
    const int lane  = tid & 31;
    const int tweet = blockIdx.x * NWAVES + wave;

    // ---- 1) load the 30 token indices for this tweet ----
    if (lane < LTOK) sIdx[wave][lane] = x[(size_t)tweet * LTOK + lane];
    __asm__ volatile("" ::: "memory");   // keep LDS stores before reads

    // ---- 2) gather embedding rows into LDS (30 rows * 16 float4 = 480) ----
#pragma unroll
    for (int it = 0; it < 15; ++it) {
        int flat = it * 32 + lane;          // 0..479
        int row  = flat >> 4;               // 0..29
        int c4   = (flat & 15) << 2;        // 0..60 step 4
        int idx  = sIdx[wave][row];
        v4f v = *(const v4f*)(emb + (size_t)idx * EDIM + c4);
        float* dst = &wg[wave][row][c4];    // 8B aligned (stride 66 breaks 16B)
        *(v2f*)(dst)     = (v2f){v.x, v.y};
        *(v2f*)(dst + 2) = (v2f){v.z, v.w};
    }
    __asm__ volatile("" ::: "memory");

    // ---- 3) WMMA fragments for row-blocks 0 (rows 0-15) and 1 (rows 16-29) --
    // A 16x4 f32 layout: lane = M%16 + 16*(K>=2), vgpr = K&1.
    // Per lane: float2 at wg[row][kb*4 + 2*(lane>>4)].
    const int half  = lane >> 4;
    const int rl    = lane & 15;
    const int row1  = 16 + ((rl < 14) ? rl : 0);  // clamp: rows 30,31 -> row 16 (junk OK)
    const float mskf = (rl < 14) ? 1.f : 0.f;     // kills G columns 30,31 at reduce time
    v2f F0[16], F1[16];
#pragma unroll
    for (int kb = 0; kb < 16; ++kb) {
        int k = kb * 4 + half * 2;
        F0[kb] = *(const v2f*)&wg[wave][rl][k];
        F1[kb] = *(const v2f*)&wg[wave][row1][k];   // unconditional, no exec masking
    }

    // ---- 4) Gram matrix G = Wg @ Wg^T (32x32, K=64) via f32 WMMA ----
    v8f c00 = {}, c01 = {}, c10 = {}, c11 = {};
#pragma unroll
    for (int kb = 0; kb < 16; ++kb) {
        c00 = __builtin_amdgcn_wmma_f32_16x16x4_f32(false, F0[kb], false, F0[kb], (short)0, c00, false, false);
        c01 = __builtin_amdgcn_wmma_f32_16x16x4_f32(false, F0[kb], false, F1[kb], (short)0, c01, false, false);
        c10 = __builtin_amdgcn_wmma_f32_16x16x4_f32(false, F1[kb], false, F0[kb], (short)0, c10, false, false);
        c11 = __builtin_amdgcn_wmma_f32_16x16x4_f32(false, F1[kb], false, F1[kb], (short)0, c11, false, false);
    }

    // ---- 5) b[m] = rowsum(G): C layout row M = 16I + r + 8*half, col = rl ---
    // J=1 tile columns 30,31 are junk -> masked by mskf (folded into the add).
#pragma unroll
    for (int r = 0; r < 8; ++r) {
        float v = fmaf(mskf, c01[r], c00[r]);
        v += __shfl_xor(v, 1, 32); v += __shfl_xor(v, 2, 32);
        v += __shfl_xor(v, 4, 32); v += __shfl_xor(v, 8, 32);
        if (rl == 0) bArr[wave][r + 8 * half] = v;
        float w = fmaf(mskf, c11[r], c10[r]);
        w += __shfl_xor(w, 1, 32); w += __shfl_xor(w, 2, 32);
        w += __shfl_xor(w, 4, 32); w += __shfl_xor(w, 8, 32);
        if (rl == 0) bArr[wave][16 + r + 8 * half] = w;   // rows 30,31 junk, unused
    }
    __asm__ volatile("" ::: "memory");

    // ---- 6) softmax over 30 tokens (one token per lane) ----
    float bl = (lane < LTOK) ? bArr[wave][lane] : -3.0e38f;
    float mx = bl;
    mx = fmaxf(mx, __shfl_xor(mx, 1, 32));
    mx = fmaxf(mx, __shfl_xor(mx, 2, 32));
    mx = fmaxf(mx, __shfl_xor(mx, 4, 32));
    mx = fmaxf(mx, __shfl_xor(mx, 8, 32));
    mx = fmaxf(mx, __shfl_xor(mx, 16, 32));
    float ev = (lane < LTOK) ? expf(bl - mx) : 0.f;
    float s = ev;
    s += __shfl_xor(s, 1, 32);
    s += __shfl_xor(s, 2, 32);
    s += __shfl_xor(s, 4, 32);
    s += __shfl_xor(s, 8, 32);
    s += __shfl_xor(s, 16, 32);
    float cl = ev / s;

    // ---- 7) ws[e] = sum_l c[l]*Wg[l][e]; each lane owns e = 2*lane, 2*lane+1 --
    // ds_load_b64 per token, bank-conflict-free (stride 66 => banks 2*lane+2l).
    float a0 = 0.f, a1 = 0.f;
#pragma unroll
    for (int l = 0; l < LTOK; ++l) {
        float c = __shfl(cl, l, 32);              // literal lane -> v_readlane
        v2f wv = *(const v2f*)&wg[wave][l][2 * lane];
        a0 = fmaf(c, wv.x, a0);
        a1 = fmaf(c, wv.y, a1);
    }
    __asm__ volatile("" ::: "memory");
    // stage pooled vector in row 0 of this wave's tile (all reads are done)
    *(v2f*)&wg[wave][0][2 * lane] = (v2f){a0, a1};
    __syncthreads();

    // ---- 8) coalesced output: out[(p*64+e)*1024 + qb + w], b64 per thread ---
    const int p  = (blockIdx.x * NWAVES) >> 10;   // (b*T + t)
    const int qb = (blockIdx.x * NWAVES) & 1023;  // h*W + w base
    {
        int e  = tid >> 2;                        // 0..63
        int w2 = tid & 3;                         // tweet pair 0..3
        v2f o = (v2f){ wg[2 * w2][0][e], wg[2 * w2 + 1][0][e] };
        *(v2f*)&out[((size_t)p * 64 + e) * 1024 + qb + 2 * w2] = o;
    }
}

extern "C" void kernel_launch(void* const* d_in, const int* in_sizes, int n_in,
                              void* d_out, int out_size, void* d_ws, size_t ws_size,
                              hipStream_t stream) {
    const int*   x   = (const int*)d_in[0];     // (8,6,32,32,30) int32
    const float* emb = (const float*)d_in[1];   // (50000,64) f32
    float*       out = (float*)d_out;           // (8, 6*64, 32, 32) f32

    int n_tweets = in_sizes[0] / LTOK;          // 49152
    int blocks   = n_tweets / NWAVES;           // 6144
    hipLaunchKernelGGL(tweetrep_kernel, dim3(blocks), dim3(256), 0, stream,
                       x, emb, out);
}